// RadiusInteractionGraph_13657996001994
// MI455X (gfx1250) — compile-verified
//
#include <hip/hip_runtime.h>
#include <hip/hip_bf16.h>

typedef float v2f __attribute__((ext_vector_type(2)));
typedef float v8f __attribute__((ext_vector_type(8)));

#define NPTS    8192
#define KNN     32
#define NGRAPH  8
#define CUT2    100.0f
#define EDGES   (NPTS * KNN + NPTS)   // 270336
#define SENT    3.0e38f

// ---------------------------------------------------------------------------
// Kernel 0a: per-graph start offsets (batch is sorted). gstart[g] =
// lower_bound(batch, g) for g = 0..NGRAPH (gstart[NGRAPH] == n).
// ---------------------------------------------------------------------------
__global__ void graph_offsets_kernel(const int* __restrict__ batch,
                                     int* __restrict__ gstart, int n) {
    int g = threadIdx.x;
    if (g > NGRAPH) return;
    int lo = 0, hi = n;
    while (lo < hi) {
        int mid = (lo + hi) >> 1;
        if (batch[mid] < g) lo = mid + 1; else hi = mid;
    }
    gstart[g] = lo;
}

// ---------------------------------------------------------------------------
// Kernel 0b: squared norms x2[i] = |pos_i|^2
// ---------------------------------------------------------------------------
__global__ void x2_kernel(const float* __restrict__ pos,
                          float* __restrict__ x2, int n) {
    int i = blockIdx.x * blockDim.x + threadIdx.x;
    if (i < n) {
        float x = pos[3 * i + 0], y = pos[3 * i + 1], z = pos[3 * i + 2];
        x2[i] = x * x + y * y + z * z;
    }
}

// ---------------------------------------------------------------------------
// Candidate filter + top-K insert for one 16x16 WMMA result tile.
// Lane holds center i = i0+(lane&15); candidate j = jt + v + 8*(lane>>4).
// ---------------------------------------------------------------------------
__device__ __forceinline__ void process_tile(const v8f& c, int jt, int half,
                                             int lane, int n, int ci,
                                             int jlo, int jhi, float x2i,
                                             const float* __restrict__ x2,
                                             float* __restrict__ sd2,
                                             int* __restrict__ sidx,
                                             float& worst, int& worstpos) {
    #pragma unroll
    for (int v = 0; v < 8; ++v) {
        int   j  = jt + v + 8 * half;            // candidate index
        int   jc = j < n ? j : n - 1;
        float d2 = fmaf(-2.0f, c[v], x2i + x2[jc]);
        bool valid = (j >= jlo) & (j < jhi) & (j != ci) & (d2 <= CUT2);
        if (valid && d2 < worst) {
            sd2[worstpos * 32 + lane] = d2;
            sidx[worstpos * 32 + lane] = j;
            // rescan for new worst
            float w = -1.0f; int wp = 0;
            #pragma unroll
            for (int e = 0; e < KNN; ++e) {
                float t = sd2[e * 32 + lane];
                if (t > w) { w = t; wp = e; }
            }
            worst = w; worstpos = wp;
        }
    }
}

// ---------------------------------------------------------------------------
// Main kernel: one wave (32 lanes) per tile of 16 centers.
//   B operand (4x16)  = the 16 centers  (K = x,y | z,0 split across lane halves)
//   A operand (16x4)  = 16 candidates j (same split)
//   D[M=j][N=i] = pos_j . pos_i  via v_wmma_f32_16x16x4_f32
// Sweep is unrolled 2x: two WMMAs per iteration share the register-resident B.
// ---------------------------------------------------------------------------
__global__ void radius_graph_kernel(const float* __restrict__ pos,
                                    const int* __restrict__ batch,
                                    const int* __restrict__ gstart,
                                    const float* __restrict__ x2,
                                    float* __restrict__ out, int n) {
    __shared__ float sd2[KNN * 32];   // [entry][lane]
    __shared__ int   sidx[KNN * 32];

    const int lane = threadIdx.x;       // 0..31
    const int half = lane >> 4;         // 0 or 1
    const int l16  = lane & 15;
    const int i0   = blockIdx.x * 16;

    // init per-lane top-K store with sentinels
    #pragma unroll
    for (int e = 0; e < KNN; ++e) {
        sd2[e * 32 + lane] = SENT;
        sidx[e * 32 + lane] = 0;
    }

    // ---- B operand: this lane's center column ----
    const int ci = i0 + l16;
    const float cx = pos[ci * 3 + 0];
    const float cy = pos[ci * 3 + 1];
    const float cz = pos[ci * 3 + 2];
    v2f bop;
    bop.x = half ? cz : cx;
    bop.y = half ? 0.0f : cy;

    const float x2i = x2[ci];
    const int   bi  = batch[ci];
    const int   jlo = gstart[bi];
    const int   jhi = gstart[bi + 1];

    // uniform sweep range covering all batches present in this center tile
    const int b_lo = batch[i0];
    const int b_hi = batch[i0 + 15];
    const int sweep_lo = gstart[b_lo] & ~15;
    const int sweep_hi = gstart[b_hi + 1];

    float worst = SENT;
    int   worstpos = 0;

    // 2x-unrolled sweep: tiles jt and jt+16. The second tile may run past
    // sweep_hi; those candidates fail j<jhi and the loads are clamped.
    for (int jt = sweep_lo; jt < sweep_hi; jt += 32) {
        // near-cache prefetch of the next iteration's two tiles
        {
            int p0 = jt + 32 + l16;      p0 = p0 < n ? p0 : n - 1;
            int p1 = jt + 48 + l16;      p1 = p1 < n ? p1 : n - 1;
            __builtin_prefetch(pos + p0 * 3, 0, 3);
            __builtin_prefetch(pos + p1 * 3, 0, 3);
        }

        // ---- A operands: candidate row M = l16 of tiles jt, jt+16 ----
        int aj0 = jt + l16;       int ajc0 = aj0 < n ? aj0 : n - 1;
        int aj1 = jt + 16 + l16;  int ajc1 = aj1 < n ? aj1 : n - 1;
        float a0x = pos[ajc0 * 3 + 0], a0y = pos[ajc0 * 3 + 1], a0z = pos[ajc0 * 3 + 2];
        float a1x = pos[ajc1 * 3 + 0], a1y = pos[ajc1 * 3 + 1], a1z = pos[ajc1 * 3 + 2];
        v2f aop0, aop1;
        aop0.x = half ? a0z : a0x;  aop0.y = half ? 0.0f : a0y;
        aop1.x = half ? a1z : a1x;  aop1.y = half ? 0.0f : a1y;

        v8f c0 = {}, c1 = {};
        c0 = __builtin_amdgcn_wmma_f32_16x16x4_f32(
                false, aop0, false, bop, (short)0, c0, false, false);
        c1 = __builtin_amdgcn_wmma_f32_16x16x4_f32(
                false, aop1, false, bop, (short)0, c1, false, false);

        process_tile(c0, jt,      half, lane, n, ci, jlo, jhi, x2i, x2,
                     sd2, sidx, worst, worstpos);
        process_tile(c1, jt + 16, half, lane, n, ci, jlo, jhi, x2i, x2,
                     sd2, sidx, worst, worstpos);
    }

    __syncthreads();

    // ---- final phase: lanes 0..15 merge the two half-streams of center i ----
    if (lane < 16) {
        const int i = i0 + lane;
        const float px = pos[i * 3 + 0];
        const float py = pos[i * 3 + 1];
        const float pz = pos[i * 3 + 2];

        for (int k = 0; k < KNN; ++k) {
            float best = SENT;
            int bestaddr = -1;
            for (int e = 0; e < 2 * KNN; ++e) {
                int addr = (e & (KNN - 1)) * 32 + lane + ((e >> 5) << 4);
                float t = sd2[addr];
                if (t < best) { best = t; bestaddr = addr; }
            }
            int   jout = 0;
            float wout = 0.0f, mout = 0.0f;
            if (bestaddr >= 0 && best < 1.0e38f) {
                jout = sidx[bestaddr];
                sd2[bestaddr] = SENT;               // remove from pool
                float dx = pos[jout * 3 + 0] - px;
                float dy = pos[jout * 3 + 1] - py;
                float dz = pos[jout * 3 + 2] - pz;
                float sq = dx * dx + dy * dy + dz * dz;
                wout = sq > 0.0f ? sqrtf(sq) : 0.0f;
                mout = 1.0f;
            }
            int base = i * KNN + k;
            out[0 * EDGES + base] = (float)jout;    // row
            out[1 * EDGES + base] = (float)i;       // col
            out[2 * EDGES + base] = wout;           // weight
            out[3 * EDGES + base] = mout;           // mask
        }
        // self-loop entry
        int e = NPTS * KNN + i;
        out[0 * EDGES + e] = (float)i;
        out[1 * EDGES + e] = (float)i;
        out[2 * EDGES + e] = 0.0f;
        out[3 * EDGES + e] = 1.0f;
    }
}

// ---------------------------------------------------------------------------
extern "C" void kernel_launch(void* const* d_in, const int* in_sizes, int n_in,
                              void* d_out, int out_size, void* d_ws, size_t ws_size,
                              hipStream_t stream) {
    const float* pos   = (const float*)d_in[0];
    const int*   batch = (const int*)d_in[1];
    const int n = in_sizes[1];           // 8192

    int*   gstart = (int*)d_ws;                          // 9 ints (padded to 64B)
    float* x2     = (float*)((char*)d_ws + 64);          // n floats
    float* out    = (float*)d_out;

    hipLaunchKernelGGL(graph_offsets_kernel, dim3(1), dim3(16), 0, stream,
                       batch, gstart, n);
    hipLaunchKernelGGL(x2_kernel, dim3((n + 255) / 256), dim3(256), 0, stream,
                       pos, x2, n);
    hipLaunchKernelGGL(radius_graph_kernel, dim3(n / 16), dim3(32), 0, stream,
                       pos, batch, gstart, x2, out, n);
}